// AgentPredictor_47991964565615
// MI455X (gfx1250) — compile-verified
//
#include <hip/hip_runtime.h>
#include <cstdint>

typedef __attribute__((ext_vector_type(2))) float v2f;
typedef __attribute__((ext_vector_type(8))) float v8f;
typedef __attribute__((ext_vector_type(4))) unsigned int v4u;
typedef __attribute__((ext_vector_type(4))) int v4i;
typedef __attribute__((ext_vector_type(8))) int v8i;

#define WAVES 4
#define SPW   16   // scenes per wave
#define NOTH  16
#define DIM   8
#define HID   16

// per-wave LDS layout (float offsets)
#define O_OTH  0      // 16 scenes * 16 others * 2 feats = 512
#define O_FOC  512    // 16 * 2                          = 32
#define O_FRU  544    // 16                              = 16
#define O_OEMB 560    // [scene][other][dim] 16*16*8     = 2048
#define O_FEMB 2608   // [scene][16] (cols 0-7 valid)    = 256
#define O_QRY  2864   // [scene][16] (cols 0-7 valid)    = 256
#define O_CTX  3120   // partials [half][scene][8]       = 256
#define O_DEC  3376   // dec_in padded [scene][12]       = 192
#define O_HBF  3568   // hidden [scene][16]              = 256
#define WSTRIDE 3824  // floats per wave (15296 B, 16B aligned)

// Wave-private LDS exchange: per-wave LDS ops execute in order (ISA 7.3),
// so only a compiler reordering fence is required, not s_barrier.
#define WAVE_LDS_FENCE() asm volatile("" ::: "memory")

__device__ __forceinline__ float relu1(float x) {
  return __builtin_amdgcn_fmed3f(x, 0.f, __builtin_inff());  // 1x v_med3_f32
}

__device__ __forceinline__ v8f wmma4(v2f a, v2f b, v8f c) {
  // D = A(16x4) * B(4x16) + C(16x16), fp32
  return __builtin_amdgcn_wmma_f32_16x16x4_f32(false, a, false, b, (short)0, c,
                                               false, false);
}

__global__ __launch_bounds__(WAVES * 32) void agent_predictor_kernel(
    const float* __restrict__ fruit, const float* __restrict__ focal,
    const float* __restrict__ others,
    const float* __restrict__ Wenc, const float* __restrict__ benc_g,
    const float* __restrict__ Wq,   const float* __restrict__ bq_g,
    const float* __restrict__ Wd1,  const float* __restrict__ bd1_g,
    const float* __restrict__ Wd2,  const float* __restrict__ bd2_g,
    float* __restrict__ out) {
  __shared__ __align__(16) float smem[WAVES * WSTRIDE];

  const int tid  = threadIdx.x;
  const int wave = tid >> 5;
  const int lane = tid & 31;
  const int n    = lane & 15;  // WMMA column (B/C/D) / row M (A) / scene id
  const int hf   = lane >> 4;  // lane half
  const int kb   = hf * 2;     // K rows carried by this half (f32 WMMA layout)
  const int WB   = wave * WSTRIDE;
  const int sceneBase = (blockIdx.x * WAVES + wave) * SPW;

  // ------------- Phase 0: stage inputs into LDS (TDM + async) -------------
#if __has_builtin(__builtin_amdgcn_tensor_load_to_lds)
  {
    // others: 16 scenes * 32 f32 = 2048 B contiguous -> 1-D TDM descriptor
    uint64_t ga = (uint64_t)(uintptr_t)(others + (size_t)sceneBase * 32);
    uint32_t la = (uint32_t)((WB + O_OTH) * 4);
    v4u g0;
    g0[0] = 1u;                                   // count=1, user, no gather
    g0[1] = la;                                   // lds_addr (bytes)
    g0[2] = (unsigned)(ga & 0xffffffffu);         // global_addr[31:0]
    g0[3] = (unsigned)((ga >> 32) & 0x01ffffffu)  // global_addr[56:32]
            | (2u << 30);                         // type = 2 (image)
    v8i g1;
    g1[0] = (int)(2u << 16);    // data_size=4B; workgroup_mask=0; no iterate/pad
    g1[1] = (int)(512u << 16);  // tensor_dim0[15:0] = 512 elements
    g1[2] = 0;                  // tensor_dim0 hi, tensor_dim1 lo
    g1[3] = (int)(512u << 16);  // tile_dim0 = 512 (bits 127:112)
    g1[4] = 0;                  // tile_dim1 = tile_dim2 = 0 (unused)
    g1[5] = 512;                // tensor_dim0_stride = 512
    g1[6] = 0;
    g1[7] = 0;
    v4i g2;  // dims 2/3 neutral: tensor_dim2=1, tensor_dim3=1, tile_dim3=0
    g2[0] = 1; g2[1] = 1; g2[2] = 512; g2[3] = 0;
    v4i g3;  // tensor_dim3_stride=512, tensor_dim4=1, tile_dim4=0
    g3[0] = 512; g3[1] = (1 << 16); g3[2] = 0; g3[3] = 0;
#if __clang_major__ >= 23
    v8i gx = {0, 0, 0, 0, 0, 0, 0, 0};
    __builtin_amdgcn_tensor_load_to_lds(g0, g1, g2, g3, gx, 0);
#else
    __builtin_amdgcn_tensor_load_to_lds(g0, g1, g2, g3, 0);
#endif
  }
#else
  {  // fallback: async global->LDS copies of the others block
    const uint64_t gsrc = (uint64_t)(uintptr_t)(others + (size_t)sceneBase * 32);
    const uint32_t lbase = (uint32_t)((WB + O_OTH) * 4);
#pragma unroll
    for (int j = 0; j < 4; ++j) {
      uint64_t ga = gsrc + (uint64_t)((lane + 32 * j) * 16);
      uint32_t la = lbase + (uint32_t)((lane + 32 * j) * 16);
      asm volatile("global_load_async_to_lds_b128 %0, %1, off"
                   :: "v"(la), "v"(ga) : "memory");
    }
  }
#endif
  {  // focal: 128 B
    uint64_t ga = (uint64_t)(uintptr_t)(focal + (size_t)sceneBase * 2) +
                  (uint64_t)(lane * 4);
    uint32_t la = (uint32_t)((WB + O_FOC) * 4 + lane * 4);
    asm volatile("global_load_async_to_lds_b32 %0, %1, off"
                 :: "v"(la), "v"(ga) : "memory");
  }
  if (lane < 16) {  // fruit: 64 B
    uint64_t ga = (uint64_t)(uintptr_t)(fruit + (size_t)sceneBase) +
                  (uint64_t)(lane * 4);
    uint32_t la = (uint32_t)((WB + O_FRU) * 4 + lane * 4);
    asm volatile("global_load_async_to_lds_b32 %0, %1, off"
                 :: "v"(la), "v"(ga) : "memory");
  }
#if __has_builtin(__builtin_amdgcn_tensor_load_to_lds)
  __builtin_amdgcn_s_wait_tensorcnt(0);
#endif
  asm volatile("s_wait_asynccnt 0" ::: "memory");
  WAVE_LDS_FENCE();

  // ------------- weight operands in registers (B layout: K in {vgpr, half}) --
  const int nm = n & 7;
  // Paired-agent encoder B: cols 0-7 <- K rows 0,1; cols 8-15 <- K rows 2,3.
  const bool act = ((hf != 0) == ((n & 8) != 0));
  v2f Benc;
  Benc.x = act ? Wenc[nm] : 0.f;
  Benc.y = act ? Wenc[DIM + nm] : 0.f;
  v2f Bq0, Bq1;  // W_q 8x8 -> two 4x16 slabs, cols>=8 zero
  Bq0.x = (n < DIM) ? Wq[(kb + 0) * DIM + n] : 0.f;
  Bq0.y = (n < DIM) ? Wq[(kb + 1) * DIM + n] : 0.f;
  Bq1.x = (n < DIM) ? Wq[(4 + kb) * DIM + n] : 0.f;
  Bq1.y = (n < DIM) ? Wq[(5 + kb) * DIM + n] : 0.f;
  v2f Bd[3];  // W_d1 11x16 padded to 12x16 -> three 4x16 slabs
#pragma unroll
  for (int t = 0; t < 3; ++t) {
    int k0 = t * 4 + kb;
    Bd[t].x = (k0 + 0 < 11) ? Wd1[(k0 + 0) * HID + n] : 0.f;
    Bd[t].y = (k0 + 1 < 11) ? Wd1[(k0 + 1) * HID + n] : 0.f;
  }
  const float bev  = benc_g[nm];              // bias per col (both agents)
  const float bqv  = (n < DIM) ? bq_g[n] : 0.f;
  const float bd1v = bd1_g[n];
  v8f Cenc, Cq, Cd1;
#pragma unroll
  for (int i = 0; i < 8; ++i) { Cenc[i] = bev; Cq[i] = bqv; Cd1[i] = bd1v; }
  float wd2r[16];
#pragma unroll
  for (int i = 0; i < 16; ++i) wd2r[i] = Wd2[i];
  const float bd2v = bd2_g[0];

  // ------------- Phase 1: shared encoder, 8 paired tiles + focal ----------
  // pair p: A K-rows = feats of others (2p, 2p+1); D col n -> other 2p+(n>=8),
  // dim n&7.  Store addr collapses to row*128 + p*16 + n (unpredicated).
#pragma unroll
  for (int p = 0; p < 8; ++p) {
    v2f a = *(const v2f*)&smem[WB + O_OTH + n * 32 + p * 4 + kb];
    v8f d = wmma4(a, Benc, Cenc);
#pragma unroll
    for (int i = 0; i < 8; ++i) d[i] = relu1(d[i]);
#pragma unroll
    for (int i = 0; i < 8; ++i)
      smem[WB + O_OEMB + (hf * 8 + i) * (NOTH * DIM) + p * 16 + n] = d[i];
  }
  {  // focal: cols 0-7 valid (cols 8-15 garbage, never read)
    v2f a = *(const v2f*)&smem[WB + O_FOC + n * 2 + kb];
    v8f d = wmma4(a, Benc, Cenc);
#pragma unroll
    for (int i = 0; i < 8; ++i) d[i] = relu1(d[i]);
#pragma unroll
    for (int i = 0; i < 8; ++i)
      smem[WB + O_FEMB + (hf * 8 + i) * 16 + n] = d[i];
  }
  WAVE_LDS_FENCE();

  // ------------- Phase 2: query = femb @ Wq + bq (K=8, 2 chained WMMA) ----
  {
    v2f a0 = *(const v2f*)&smem[WB + O_FEMB + n * 16 + 0 + kb];
    v2f a1 = *(const v2f*)&smem[WB + O_FEMB + n * 16 + 4 + kb];
    v8f d = wmma4(a0, Bq0, Cq);
    d = wmma4(a1, Bq1, d);
#pragma unroll
    for (int i = 0; i < 8; ++i)
      smem[WB + O_QRY + (hf * 8 + i) * 16 + n] = d[i];
  }
  WAVE_LDS_FENCE();

  // ------------- Phase 3: attention (VALU). lane = (scene=n, half=hf) -----
  const int s  = n;
  const int n0 = hf * 8;  // this half handles others n0..n0+7
  float wgt[8];
  {
    const float* qr = &smem[WB + O_QRY + s * 16];
    float q[8];
#pragma unroll
    for (int i = 0; i < 8; ++i) q[i] = qr[i];
    const float rs = 0.35355339059327373f;  // 1/sqrt(8)
    float sc[8];
#pragma unroll
    for (int j = 0; j < 8; ++j) {
      const float* er = &smem[WB + O_OEMB + s * (NOTH * DIM) + (n0 + j) * DIM];
      float acc = 0.f;
#pragma unroll
      for (int dd = 0; dd < 8; ++dd) acc += er[dd] * q[dd];
      sc[j] = acc * rs;
    }
    // softmax over 16: local 8 + cross-half exchange (no LDS round-trip)
    float mx = sc[0];
#pragma unroll
    for (int j = 1; j < 8; ++j) mx = fmaxf(mx, sc[j]);
    mx = fmaxf(mx, __shfl_xor(mx, 16, 32));
    float ssum = 0.f;
#pragma unroll
    for (int j = 0; j < 8; ++j) {
      float e = __expf(sc[j] - mx);
      wgt[j] = e;
      ssum += e;
    }
    ssum += __shfl_xor(ssum, 16, 32);
    float inv = 1.f / ssum;
#pragma unroll
    for (int j = 0; j < 8; ++j) wgt[j] *= inv;
  }
  {  // context partial over this half's 8 others
    float acc[8];
#pragma unroll
    for (int dd = 0; dd < 8; ++dd) acc[dd] = 0.f;
#pragma unroll
    for (int j = 0; j < 8; ++j) {
      const float* er = &smem[WB + O_OEMB + s * (NOTH * DIM) + (n0 + j) * DIM];
#pragma unroll
      for (int dd = 0; dd < 8; ++dd) acc[dd] += wgt[j] * er[dd];
    }
#pragma unroll
    for (int dd = 0; dd < 8; ++dd)
      smem[WB + O_CTX + (hf * 16 + s) * 8 + dd] = acc[dd];
  }
  WAVE_LDS_FENCE();

  // ------------- Phase 4: dec_in = [fruit, focal, ctx] padded to 12 -------
  if (hf == 0) {
    smem[WB + O_DEC + s * 12 + 0] = smem[WB + O_FRU + s];
    smem[WB + O_DEC + s * 12 + 1] = smem[WB + O_FOC + s * 2 + 0];
    smem[WB + O_DEC + s * 12 + 2] = smem[WB + O_FOC + s * 2 + 1];
#pragma unroll
    for (int dd = 0; dd < 3; ++dd)
      smem[WB + O_DEC + s * 12 + 3 + dd] =
          smem[WB + O_CTX + s * 8 + dd] + smem[WB + O_CTX + (16 + s) * 8 + dd];
  } else {
#pragma unroll
    for (int dd = 3; dd < 8; ++dd)
      smem[WB + O_DEC + s * 12 + 3 + dd] =
          smem[WB + O_CTX + s * 8 + dd] + smem[WB + O_CTX + (16 + s) * 8 + dd];
    smem[WB + O_DEC + s * 12 + 11] = 0.f;
  }
  WAVE_LDS_FENCE();

  // ------------- Phase 5: hidden = relu(dec_in @ Wd1 + bd1), K=12 ---------
  {
    v2f a0 = *(const v2f*)&smem[WB + O_DEC + n * 12 + 0 + kb];
    v2f a1 = *(const v2f*)&smem[WB + O_DEC + n * 12 + 4 + kb];
    v2f a2 = *(const v2f*)&smem[WB + O_DEC + n * 12 + 8 + kb];
    v8f d = wmma4(a0, Bd[0], Cd1);
    d = wmma4(a1, Bd[1], d);
    d = wmma4(a2, Bd[2], d);
#pragma unroll
    for (int i = 0; i < 8; ++i) d[i] = relu1(d[i]);
#pragma unroll
    for (int i = 0; i < 8; ++i)
      smem[WB + O_HBF + (hf * 8 + i) * 16 + n] = d[i];
  }
  WAVE_LDS_FENCE();

  // ------------- Phase 6: out = sigmoid(h . Wd2 + b) ----------------------
  {
    const float* hr = &smem[WB + O_HBF + s * 16];
    float acc = bd2v;
#pragma unroll
    for (int i = 0; i < 16; ++i) acc += hr[i] * wd2r[i];
    float sg = 1.f / (1.f + __expf(-acc));
    if (hf == 0) out[sceneBase + s] = sg;
  }
}

extern "C" void kernel_launch(void* const* d_in, const int* in_sizes, int n_in,
                              void* d_out, int out_size, void* d_ws,
                              size_t ws_size, hipStream_t stream) {
  const float* fruit  = (const float*)d_in[0];
  const float* focal  = (const float*)d_in[1];
  const float* others = (const float*)d_in[2];
  const float* Wenc = (const float*)d_in[3];
  const float* benc = (const float*)d_in[4];
  const float* Wq   = (const float*)d_in[5];
  const float* bq   = (const float*)d_in[6];
  const float* Wd1  = (const float*)d_in[7];
  const float* bd1  = (const float*)d_in[8];
  const float* Wd2  = (const float*)d_in[9];
  const float* bd2  = (const float*)d_in[10];

  const int B = in_sizes[0];             // 1048576, divisible by 64
  const int blocks = B / (WAVES * SPW);  // 16384
  agent_predictor_kernel<<<blocks, WAVES * 32, 0, stream>>>(
      fruit, focal, others, Wenc, benc, Wq, bq, Wd1, bd1, Wd2, bd2,
      (float*)d_out);
}